// DistMaps_28217935135424
// MI455X (gfx1250) — compile-verified
//
#include <hip/hip_runtime.h>
#include <hip/hip_bf16.h>

typedef __attribute__((ext_vector_type(2))) float v2f;
typedef __attribute__((ext_vector_type(8))) float v8f;

#define H_ 512
#define W_ 512
#define NPTS 12
#define TILE 16
#define T_COLS 2               // column-tiles per wave (A operands reused)
#define GROUPS_X (W_ / TILE / T_COLS)                 // 16
#define GROUPS_PER_IMG ((H_ / TILE) * GROUPS_X)       // 512
#define BLOCKS_PER_IMG (GROUPS_PER_IMG / 8)           // 64 (8 waves/block)
#define INV_SCALE 0.2f         // 1 / (NORM_RADIUS * SPATIAL_SCALE)
#define INVALID_DIST 1000000.0f

__device__ __forceinline__ float dm_tanh(float z) {
#if __has_builtin(__builtin_amdgcn_tanhf)
  return __builtin_amdgcn_tanhf(z);           // gfx1250 v_tanh_f32
#else
  // z >= 0 here; tanh(z) = 1 - 2/(e^{2z}+1), e^{2z} = exp2(2z*log2e)
  float e = __builtin_amdgcn_exp2f(z * (2.0f * 1.4426950408889634f));
  return 1.0f - 2.0f / (e + 1.0f);
#endif
}

__global__ __launch_bounds__(256)
void distmaps_wmma_kernel(const float* __restrict__ coords,
                          float* __restrict__ out) {
  const int lane  = threadIdx.x & 31;
  const int wave  = threadIdx.x >> 5;
  const int laneM = lane & 15;
  const bool lo   = lane < 16;

  // img is a pure function of blockIdx -> coords address is scalar-uniform
  const int img     = blockIdx.x >> 6;             // / BLOCKS_PER_IMG
  const int rem_blk = blockIdx.x & (BLOCKS_PER_IMG - 1);
  const int rem     = rem_blk * 8 + wave;          // 0..511 (wave-uniform)
  const int tile_r  = rem >> 4;                    // / GROUPS_X
  const int group_c = rem & (GROUPS_X - 1);
  const int row_base  = tile_r * TILE;
  const int col_base0 = group_c * (T_COLS * TILE);

  const int b = img >> 1, g = img & 1;
  const float* cp = coords + ((size_t)b * 24 + (size_t)g * NPTS) * 3;

  // ---- Build A-matrix operand column (K0) per point: dr2 for lanes 0-15 ----
  float a0[NPTS];   // VGPR0 of A: K=0 (lanes 0-15), K=2 (lanes 16-31, zero)
  float xs[NPTS];
  const float fr = (float)(row_base + laneM);
  #pragma unroll
  for (int p = 0; p < NPTS; ++p) {
    float y = cp[p * 3 + 0];                       // uniform -> s_load
    float x = cp[p * 3 + 1];
    xs[p] = x;
    float dy  = (fr - y) * INV_SCALE;
    float dr2 = dy * dy;
    bool invalid = fmaxf(y, x) < 0.0f;
    dr2 = invalid ? INVALID_DIST : dr2;            // 1e6 dominates any valid d
    a0[p] = lo ? dr2 : 0.0f;
  }
  const float a1    = lo ? 1.0f : 0.0f;            // A column K=1 = ones
  const float brow0 = a1;                          // B row K=0 = ones

  const size_t img_off = (size_t)img * H_ * W_;

  #pragma unroll
  for (int t = 0; t < T_COLS; ++t) {
    const int col_base = col_base0 + t * TILE;
    const float fc = (float)(col_base + laneM);

    v8f cmin;
    #pragma unroll
    for (int p = 0; p < NPTS; ++p) {
      float dx  = (fc - xs[p]) * INV_SCALE;
      float dc2 = dx * dx;
      v2f A, Bm;
      A[0]  = a0[p];            // K0: dr2(row) | K2: 0
      A[1]  = a1;               // K1: 1        | K3: 0
      Bm[0] = brow0;            // K0 row: 1    | K2 row: 0
      Bm[1] = lo ? dc2 : 0.0f;  // K1 row: dc2  | K3 row: 0
      v8f zero = {};
      // D(i,j) = dr2_p(i) + dc2_p(j) : rank-2 outer sum as one WMMA
      v8f d = __builtin_amdgcn_wmma_f32_16x16x4_f32(
          false, A, false, Bm, (short)0, zero, false, false);
      if (p == 0) {
        cmin = d;
      } else {
        #pragma unroll
        for (int k = 0; k < 8; ++k) cmin[k] = __builtin_fminf(cmin[k], d[k]);
      }
    }

    // D layout: VGPR k -> row (k | k+8), col = lane&15
    size_t base = img_off + (size_t)(row_base + (lo ? 0 : 8)) * W_
                + (size_t)col_base + laneM;
    #pragma unroll
    for (int k = 0; k < 8; ++k) {
      float s = __builtin_amdgcn_sqrtf(cmin[k]);
      out[base + (size_t)k * W_] = dm_tanh(2.0f * s);
    }
  }
}

extern "C" void kernel_launch(void* const* d_in, const int* in_sizes, int n_in,
                              void* d_out, int out_size, void* d_ws, size_t ws_size,
                              hipStream_t stream) {
  // inputs: d_in[0] = x (unused, shape only), d_in[1] = coords (B,24,3) f32
  const float* coords = (const float*)d_in[1];
  float* out = (float*)d_out;

  int B    = in_sizes[1] / (24 * 3);   // 8
  int nimg = B * 2;                    // 16 output planes

  int blocks = nimg * BLOCKS_PER_IMG;  // 1024 blocks x 256 threads (8 waves)
  distmaps_wmma_kernel<<<blocks, 256, 0, stream>>>(coords, out);
}